// QuantumHybridQLSTM_65481071399265
// MI455X (gfx1250) — compile-verified
//
#include <hip/hip_runtime.h>

// MI455X / gfx1250, wave32. Latency-bound recurrent scan:
//  - batch rows independent -> 16 blocks x 16 rows, zero inter-block sync
//  - GEMM: V_WMMA_F32_16X16X4_F32 (fp32-exact), 2 gates packed per 16x16 tile
//  - quantum circuit reduced analytically to prefix products of cosines

typedef float v2f __attribute__((ext_vector_type(2)));
typedef float v8f __attribute__((ext_vector_type(8)));

#define SEQ     64
#define BATCH   256
#define IN_DIM  64
#define HID     8
#define D_IN    72      // IN_DIM + HID
#define KFRAGS  18      // 72 / 4
#define XFRAGS  16      // 64 / 4

__device__ __forceinline__ float fast_sigmoid(float x) {
    return 1.0f / (1.0f + __expf(-x));
}
__device__ __forceinline__ float fast_tanh(float x) {
    // stable: -> -1 for x<<0, +1 for x>>0
    return 1.0f - 2.0f / (1.0f + __expf(2.0f * x));
}

__global__ __launch_bounds__(64, 1)
void qlstm_wmma_kernel(const float* __restrict__ x,
                       const float* __restrict__ Wf, const float* __restrict__ bf, const float* __restrict__ qf,
                       const float* __restrict__ Wi, const float* __restrict__ bi, const float* __restrict__ qi,
                       const float* __restrict__ Wg, const float* __restrict__ bg, const float* __restrict__ qg,
                       const float* __restrict__ Wo, const float* __restrict__ bo, const float* __restrict__ qo,
                       float* __restrict__ out)
{
    __shared__ __align__(16) float theta_lds[2][16 * 16];   // per-wave theta tile
    __shared__ __align__(16) float gates_lds[4][16 * HID];  // f, i, g, o
    __shared__ __align__(16) float hx_lds[16 * HID];
    __shared__ __align__(16) float cx_lds[16 * HID];

    const int tid  = threadIdx.x;
    const int lane = tid & 31;
    const int wv   = tid >> 5;       // wave id == gate-pair index (0: f+i, 1: g+o)
    const int n16  = lane & 15;      // WMMA N column / row index
    const int hi   = lane >> 4;      // half-wave
    const int row0 = blockIdx.x * 16;

    // gate owned by this lane in the WMMA B operand (selected by N column)
    const int giB = 2 * wv + (n16 >> 3);
    // gate owned by this lane in the row-per-lane stage (selected by half-wave)
    const int giR = 2 * wv + hi;

    const float* WB = (giB == 0) ? Wf : (giB == 1) ? Wi : (giB == 2) ? Wg : Wo;
    const float* bR = (giR == 0) ? bf : (giR == 1) ? bi : (giR == 2) ? bg : bo;
    const float* qR = (giR == 0) ? qf : (giR == 1) ? qi : (giR == 2) ? qg : qo;

    // ---- preload WMMA B fragments (weights) once, keep in VGPRs ----
    // B tile is 4x16 (KxN) f32: lane n16 = N; VGPR0 = B[2*hi][N], VGPR1 = B[2*hi+1][N]
    // B[k][n] = W_gate[n&7][k]  (theta = comb @ W^T)
    v2f Bfrag[KFRAGS];
    {
        const int wrow = n16 & 7;
        #pragma unroll
        for (int kk = 0; kk < KFRAGS; ++kk) {
            const int kb = kk * 4 + 2 * hi;
            Bfrag[kk] = *(const v2f*)(WB + wrow * D_IN + kb);
        }
    }

    // ---- per-lane analytic-circuit constants: cos(q_w) for w<7, raw q_7, biases ----
    float qc[7], q7, bias[8];
    #pragma unroll
    for (int j = 0; j < 7; ++j) qc[j] = __cosf(qR[j]);
    q7 = qR[7];
    #pragma unroll
    for (int j = 0; j < 8; ++j) bias[j] = bR[j];

    for (int e = tid; e < 16 * HID; e += 64) { hx_lds[e] = 0.0f; cx_lds[e] = 0.0f; }
    __syncthreads();

    for (int t = 0; t < SEQ; ++t) {
        const float* xrow = x + ((size_t)t * BATCH + row0 + n16) * IN_DIM;

        // ---- A fragments: comb = [x_t | hx], 16 rows x 72 cols ----
        // A tile 16x4 (MxK) f32: lane n16 = M; VGPR0 = A[M][2*hi], VGPR1 = A[M][2*hi+1]
        v2f Afrag[KFRAGS];
        #pragma unroll
        for (int kk = 0; kk < XFRAGS; ++kk)
            Afrag[kk] = *(const v2f*)(xrow + kk * 4 + 2 * hi);
        #pragma unroll
        for (int kk = XFRAGS; kk < KFRAGS; ++kk)
            Afrag[kk] = *(const v2f*)(&hx_lds[n16 * HID + (kk - XFRAGS) * 4 + 2 * hi]);

        if (t + 1 < SEQ) __builtin_prefetch(xrow + BATCH * IN_DIM, 0, 1);

        // ---- theta(16x16) = comb(16x72) @ [W_glo^T | W_ghi^T](72x16), fp32 WMMA ----
        v8f acc0 = {0.f,0.f,0.f,0.f,0.f,0.f,0.f,0.f};
        v8f acc1 = {0.f,0.f,0.f,0.f,0.f,0.f,0.f,0.f};
        #pragma unroll
        for (int kk = 0; kk < KFRAGS; kk += 2) {
            acc0 = __builtin_amdgcn_wmma_f32_16x16x4_f32(
                false, Afrag[kk],     false, Bfrag[kk],     (short)0, acc0, false, false);
            acc1 = __builtin_amdgcn_wmma_f32_16x16x4_f32(
                false, Afrag[kk + 1], false, Bfrag[kk + 1], (short)0, acc1, false, false);
        }
        v8f acc = acc0 + acc1;

        // ---- dump D layout to LDS: VGPR v -> row (v + 8*hi), col n16 ----
        #pragma unroll
        for (int v = 0; v < 8; ++v)
            theta_lds[wv][(v + 8 * hi) * 16 + n16] = acc[v];

        // ---- row-per-lane quantum stage: lane = (row n16, gate giR) ----
        // <Z_w> = cos(q_w) * prod_{j<=w} cos(theta_j)   (w<7)
        // <Z_7> = cos(theta_7 + q_7) * prod_{j<7} cos(theta_j)
        float e8[8];
        {
            const float* th = &theta_lds[wv][n16 * 16 + 8 * hi];
            float p = 1.0f;
            #pragma unroll
            for (int j = 0; j < 7; ++j) {
                p *= __cosf(th[j] + bias[j]);
                e8[j] = qc[j] * p;
            }
            e8[7] = p * __cosf(th[7] + bias[7] + q7);
        }
        float gv[8];
        if (giR == 2) {                 // gate g -> tanh
            #pragma unroll
            for (int j = 0; j < 8; ++j) gv[j] = fast_tanh(e8[j]);
        } else {                        // f, i, o -> sigmoid
            #pragma unroll
            for (int j = 0; j < 8; ++j) gv[j] = fast_sigmoid(e8[j]);
        }
        #pragma unroll
        for (int j = 0; j < 8; ++j) gates_lds[giR][n16 * HID + j] = gv[j];

        __syncthreads();

        // ---- LSTM update: 128 elements / 64 threads, 2 each ----
        {
            const int e0 = tid * 2;
            const float2 f2 = *(const float2*)&gates_lds[0][e0];
            const float2 i2 = *(const float2*)&gates_lds[1][e0];
            const float2 g2 = *(const float2*)&gates_lds[2][e0];
            const float2 o2 = *(const float2*)&gates_lds[3][e0];
            const float2 c2 = *(const float2*)&cx_lds[e0];
            const float cxa = f2.x * c2.x + i2.x * g2.x;
            const float cxb = f2.y * c2.y + i2.y * g2.y;
            const float hxa = o2.x * fast_tanh(cxa);
            const float hxb = o2.y * fast_tanh(cxb);
            *(float2*)&cx_lds[e0] = make_float2(cxa, cxb);
            *(float2*)&hx_lds[e0] = make_float2(hxa, hxb);
            *(float2*)(out + ((size_t)t * BATCH + row0) * HID + e0) = make_float2(hxa, hxb);
        }
        __syncthreads();  // hx visible to both waves' A-fragment loads next step
    }

    // ---- final (hx, cx) tail outputs ----
    {
        const int e0 = tid * 2;
        const size_t obase = (size_t)SEQ * BATCH * HID;
        const float2 h2 = *(const float2*)&hx_lds[e0];
        const float2 c2 = *(const float2*)&cx_lds[e0];
        *(float2*)(out + obase + (size_t)row0 * HID + e0) = h2;
        *(float2*)(out + obase + (size_t)BATCH * HID + (size_t)row0 * HID + e0) = c2;
    }
}

extern "C" void kernel_launch(void* const* d_in, const int* in_sizes, int n_in,
                              void* d_out, int out_size, void* d_ws, size_t ws_size,
                              hipStream_t stream) {
    const float* x  = (const float*)d_in[0];
    const float* Wf = (const float*)d_in[1];
    const float* bf = (const float*)d_in[2];
    const float* qf = (const float*)d_in[3];
    const float* Wi = (const float*)d_in[4];
    const float* bi = (const float*)d_in[5];
    const float* qi = (const float*)d_in[6];
    const float* Wg = (const float*)d_in[7];
    const float* bg = (const float*)d_in[8];
    const float* qg = (const float*)d_in[9];
    const float* Wo = (const float*)d_in[10];
    const float* bo = (const float*)d_in[11];
    const float* qo = (const float*)d_in[12];
    float* out = (float*)d_out;

    qlstm_wmma_kernel<<<BATCH / 16, 64, 0, stream>>>(
        x, Wf, bf, qf, Wi, bi, qi, Wg, bg, qg, Wo, bo, qo, out);
}